// PolicyNetwork_72584947302715
// MI455X (gfx1250) — compile-verified
//
#include <hip/hip_runtime.h>
#include <hip/hip_bf16.h>
#include <math.h>

// Problem dims
#define AGV 32
#define ORD 100
#define MCH 50
#define EMB 128
#define ROWLEN (ORD * MCH + MCH)   // 5050 per-AGV logits

typedef __attribute__((ext_vector_type(2))) float v2f;
typedef __attribute__((ext_vector_type(8))) float v8f;

// ---------------------------------------------------------------------------
// Stage 1: five f32 GEMMs  [rows,128] @ [128,128] -> [rows,128]
//   pa = h_agv    @ W1[0:128]      (rows=32)
//   po = h_order  @ W1[128:256]    (rows=100)
//   pw = h_waiting@ W1[256:384]    (rows=50)
//   qa = h_agv    @ Wp1[0:128]     (rows=32)
//   qb = h_buffer @ Wp1[128:256]   (rows=50)
// One wave (32 threads) per 16x16 output tile, K in steps of 4 via
// V_WMMA_F32_16X16X4_F32 (32 WMMA issues per tile).
// Partial row tiles handled BRANCHLESSLY: clamp the row pointer to a legal
// row and scale loaded A values by 0/1 — keeps EXEC all-ones statically so
// the loads can be clause'd/hoisted ahead of the WMMA chain (no per-iter
// s_and_saveexec / s_cbranch_execz).
// Row-tiles per GEMM: 2,7,4,2,4 (prefix 0,2,9,13,15,19) x 8 col-tiles = 152.
// ---------------------------------------------------------------------------
__global__ __launch_bounds__(32)
void policy_stage1_wmma(const float* __restrict__ hOrd,
                        const float* __restrict__ hAgv,
                        const float* __restrict__ hWait,
                        const float* __restrict__ hBuf,
                        const float* __restrict__ W1,
                        const float* __restrict__ Wp1,
                        float* __restrict__ pa, float* __restrict__ po,
                        float* __restrict__ pw, float* __restrict__ qa,
                        float* __restrict__ qb) {
    const int rtPrefix[6] = {0, 2, 9, 13, 15, 19};

    int tile    = blockIdx.x;
    int colTile = tile & 7;        // 8 column tiles of 16 (EMB=128)
    int rt      = tile >> 3;       // global row-tile id 0..18

    int g = 0;
    while (rt >= rtPrefix[g + 1]) ++g;
    int rowTile = rt - rtPrefix[g];

    const float* src; const float* w; float* dst; int rows;
    switch (g) {
        case 0:  src = hAgv;  w = W1;                  dst = pa; rows = AGV; break;
        case 1:  src = hOrd;  w = W1  + EMB * EMB;     dst = po; rows = ORD; break;
        case 2:  src = hWait; w = W1  + 2 * EMB * EMB; dst = pw; rows = MCH; break;
        case 3:  src = hAgv;  w = Wp1;                 dst = qa; rows = AGV; break;
        default: src = hBuf;  w = Wp1 + EMB * EMB;     dst = qb; rows = MCH; break;
    }

    const int lane    = threadIdx.x;      // 0..31 (wave32)
    const int half    = lane >> 4;        // 0: K pair {0,1}, 1: K pair {2,3}
    const int mn      = lane & 15;        // row index for A, col index for B
    const int rowBase = rowTile * 16;
    const int colBase = colTile * 16;

    // Branchless partial-tile handling for the A operand.
    const int   r        = rowBase + mn;
    const float rowScale = (r < rows) ? 1.0f : 0.0f;        // zero-fill OOB rows
    const int   rClamp   = (r < rows) ? r : (rows - 1);     // always-legal load
    const float* __restrict__ srcRow = src + rClamp * EMB + 2 * half;
    const float* __restrict__ wCol   = w + (2 * half) * EMB + colBase + mn;

    v8f acc = {};   // 16x16 f32 accumulator, 8 VGPRs
    #pragma unroll
    for (int k0 = 0; k0 < EMB; k0 += 4) {
        // A: 16x4 tile. Lane holds src[r][k0+2*half], src[r][k0+2*half+1].
        v2f a;
        a.x = srcRow[k0]     * rowScale;
        a.y = srcRow[k0 + 1] * rowScale;

        // B: 4x16 tile. Lane holds w[k0+2*half][col], w[k0+2*half+1][col].
        v2f b;
        b.x = wCol[k0 * EMB];
        b.y = wCol[(k0 + 1) * EMB];

        // (neg_a, A, neg_b, B, c_mod, C, reuse_a, reuse_b)
        acc = __builtin_amdgcn_wmma_f32_16x16x4_f32(
                  false, a, false, b, (short)0, acc, false, false);
    }

    // D layout: VGPR i holds D[i + 8*half][mn]
    #pragma unroll
    for (int i = 0; i < 8; ++i) {
        const int rr = rowBase + i + 8 * half;
        if (rr < rows) dst[rr * EMB + colBase + mn] = acc[i];
    }
}

// ---------------------------------------------------------------------------
// Wave32 full reduction (sum) via lane shuffles.
// ---------------------------------------------------------------------------
__device__ inline float wave32_sum(float v) {
    #pragma unroll
    for (int off = 16; off > 0; off >>= 1)
        v += __shfl_xor(v, off, 32);
    return v;
}

// ---------------------------------------------------------------------------
// Stage 2: dispatch[a,o,m] = relu(pa[a]+po[o]+pw[m]+b1) . W2 + b2, masked.
// One block per (a,o); 4 waves; each wave owns a strided subset of m.
// base = pa[a]+po[o]+b1 and W2 live in registers (4 floats/lane), only
// pw[m] (25.6 KB total, L2/WGP$-resident) streams per iteration.
// ---------------------------------------------------------------------------
__global__ __launch_bounds__(128)
void policy_stage2_dispatch(const float* __restrict__ pa,
                            const float* __restrict__ po,
                            const float* __restrict__ pw,
                            const float* __restrict__ b1,
                            const float* __restrict__ W2,
                            const float* __restrict__ b2,
                            const int*   __restrict__ mask,
                            float* __restrict__ out) {
    const int a    = blockIdx.x / ORD;
    const int o    = blockIdx.x % ORD;
    const int tid  = threadIdx.x;
    const int wave = tid >> 5;
    const int lane = tid & 31;

    float base[4], w2v[4];
    #pragma unroll
    for (int j = 0; j < 4; ++j) {
        const int e = lane + 32 * j;
        base[j] = pa[a * EMB + e] + po[o * EMB + e] + b1[e];
        w2v[j]  = W2[e];
    }
    const float b2s  = b2[0];
    const float ninf = -__builtin_huge_valf();

    for (int m = wave; m < MCH; m += 4) {
        float s = 0.0f;
        #pragma unroll
        for (int j = 0; j < 4; ++j) {
            float x = base[j] + pw[m * EMB + lane + 32 * j];
            x = x > 0.0f ? x : 0.0f;
            s = fmaf(x, w2v[j], s);
        }
        s = wave32_sum(s);
        if (lane == 0) {
            const int idx = a * ROWLEN + o * MCH + m;
            out[idx] = (mask[idx] == 0) ? ninf : (s + b2s);
        }
    }
}

// ---------------------------------------------------------------------------
// Stage 3: pickup[a,m] = relu(qa[a]+qb[m]+bp1) . Wp2 + bp2, masked.
// One wave per (a,m); 400 blocks x 4 waves = 1600 outputs.
// ---------------------------------------------------------------------------
__global__ __launch_bounds__(128)
void policy_stage3_pickup(const float* __restrict__ qa,
                          const float* __restrict__ qb,
                          const float* __restrict__ bp1,
                          const float* __restrict__ Wp2,
                          const float* __restrict__ bp2,
                          const int*   __restrict__ mask,
                          float* __restrict__ out) {
    const int tid  = threadIdx.x;
    const int wave = tid >> 5;
    const int lane = tid & 31;
    const int wg   = blockIdx.x * 4 + wave;   // 0..1599
    if (wg >= AGV * MCH) return;
    const int a = wg / MCH;
    const int m = wg % MCH;

    float s = 0.0f;
    #pragma unroll
    for (int j = 0; j < 4; ++j) {
        const int e = lane + 32 * j;
        float x = qa[a * EMB + e] + qb[m * EMB + e] + bp1[e];
        x = x > 0.0f ? x : 0.0f;
        s = fmaf(x, Wp2[e], s);
    }
    s = wave32_sum(s);
    if (lane == 0) {
        const int idx = a * ROWLEN + ORD * MCH + m;
        out[idx] = (mask[idx] == 0) ? -__builtin_huge_valf() : (s + bp2[0]);
    }
}

// ---------------------------------------------------------------------------
// Host launcher
// ---------------------------------------------------------------------------
extern "C" void kernel_launch(void* const* d_in, const int* in_sizes, int n_in,
                              void* d_out, int out_size, void* d_ws, size_t ws_size,
                              hipStream_t stream) {
    const float* hOrd  = (const float*)d_in[0];   // [O,E]
    const float* hAgv  = (const float*)d_in[1];   // [A,E]
    const float* hWait = (const float*)d_in[2];   // [M,E]
    const float* hBuf  = (const float*)d_in[3];   // [M,E]
    const float* W1    = (const float*)d_in[4];   // [3E,E]
    const float* b1    = (const float*)d_in[5];   // [E]
    const float* W2    = (const float*)d_in[6];   // [E,1]
    const float* b2    = (const float*)d_in[7];   // [1]
    const float* Wp1   = (const float*)d_in[8];   // [2E,E]
    const float* bp1   = (const float*)d_in[9];   // [E]
    const float* Wp2   = (const float*)d_in[10];  // [E,1]
    const float* bp2   = (const float*)d_in[11];  // [1]
    const int*   amask = (const int*)d_in[12];    // [A*(O*M+M)]
    float* out = (float*)d_out;

    // Workspace partition (floats): pa, po, pw, qa, qb (264*128 floats = 132 KB)
    float* pa = (float*)d_ws;
    float* po = pa + AGV * EMB;
    float* pw = po + ORD * EMB;
    float* qa = pw + MCH * EMB;
    float* qb = qa + AGV * EMB;

    // Stage 1: 152 tiles, one wave each.
    policy_stage1_wmma<<<152, 32, 0, stream>>>(hOrd, hAgv, hWait, hBuf, W1, Wp1,
                                               pa, po, pw, qa, qb);

    // Stage 2: one block per (a,o) pair.
    policy_stage2_dispatch<<<AGV * ORD, 128, 0, stream>>>(pa, po, pw, b1, W2, b2,
                                                          amask, out);

    // Stage 3: 1600 (a,m) waves across 400 blocks.
    policy_stage3_pickup<<<(AGV * MCH + 3) / 4, 128, 0, stream>>>(qa, qb, bp1, Wp2,
                                                                  bp2, amask, out);
}